// DoRAEmbedding_2018634629740
// MI455X (gfx1250) — compile-verified
//
#include <hip/hip_runtime.h>
#include <hip/hip_bf16.h>
#include <math.h>

// DoRA embedding, MI455X (gfx1250, wave32).
// Phase 1: column sum-of-squares of combined = E + 2*(A@B) over full 16-row
//          tiles only (no predication in hot loop), WMMA f32 16x16x4.
// Phase 2: finalize adds the single tail row, computes mag/max(norm,1e-8).
// Phase 3: gather 16384 token rows, lora tile via WMMA, scale, NT-store.

#define VOCAB 50257
#define DIM   768
#define RANK  16
#define SCALING 2.0f               // alpha/rank = 32/16
#define NVF  (VOCAB / 16)          // 3141 full v-tiles; tail row 50256 in finalize

typedef float v2f __attribute__((ext_vector_type(2)));
typedef float v8f __attribute__((ext_vector_type(8)));

// ---------------------------------------------------------------------------
// Phase 1: sumsq[d] += sum over full tiles of (emb + SCALING*(A@B))^2
// Block = 256 threads = 8 waves; wave w owns d-tiles {6w..6w+5}.
// V tiles grid-strided across blocks. Register accumulation across v-tiles;
// one LDS atomic per (wave, dtile); one global atomic per column per block.
// ---------------------------------------------------------------------------
__global__ void __launch_bounds__(256) dora_colnorm_kernel(
    const float* __restrict__ emb,
    const float* __restrict__ la,   // [VOCAB, 16]
    const float* __restrict__ lb,   // [16, DIM]
    float* __restrict__ sumsq)      // [DIM], pre-zeroed
{
  __shared__ float lds_sumsq[DIM];
  const int tid = threadIdx.x;
  for (int d = tid; d < DIM; d += 256) lds_sumsq[d] = 0.0f;
  __syncthreads();

  const int lane = tid & 31;
  const int wave = tid >> 5;
  const int half = lane >> 4;     // 0: lanes 0-15, 1: lanes 16-31
  const int l16  = lane & 15;

  const float2* la2 = reinterpret_cast<const float2*>(la); // row = 8 float2
  const long vstep = (long)gridDim.x * 16;                 // rows per stride

  for (int i = 0; i < 6; ++i) {
    const int dtile = wave * 6 + i;
    const int d0 = dtile * 16;

    // B chunk j: K rows {4j,4j+1} (lanes 0-15) / {4j+2,4j+3} (lanes 16-31),
    // column N = d0 + l16. Loaded once per dtile, reused for all v-tiles.
    v2f b[4];
#pragma unroll
    for (int j = 0; j < 4; ++j) {
      const int k = 4 * j + 2 * half;
      b[j][0] = lb[(k    ) * DIM + d0 + l16];
      b[j][1] = lb[(k + 1) * DIM + d0 + l16];
    }

    // Incremented base pointers: A row M = l16, emb rows (8*half + r).
    const float2* ap = la2 + ((long)blockIdx.x * 16 + l16) * 8 + half;
    const float*  ep = emb + ((long)blockIdx.x * 16 + 8 * half) * DIM + d0 + l16;

    float part = 0.0f;
    for (int vt = blockIdx.x; vt < NVF; vt += gridDim.x) {
      v2f a[4];
#pragma unroll
      for (int j = 0; j < 4; ++j) {          // 4x global_load_b64, imm offsets
        const float2 t = ap[2 * j];
        a[j][0] = t.x; a[j][1] = t.y;
      }
      float e[8];
#pragma unroll
      for (int r = 0; r < 8; ++r)            // 8x global_load_b32, imm offsets
        e[r] = ep[r * DIM];

      v8f c = {};
#pragma unroll
      for (int j = 0; j < 4; ++j)
        c = __builtin_amdgcn_wmma_f32_16x16x4_f32(
                false, a[j], false, b[j], (short)0, c, false, false);

#pragma unroll
      for (int r = 0; r < 8; ++r) {          // D: VGPR r -> M = r+8*half, N = l16
        const float cm = fmaf(SCALING, c[r], e[r]);
        part = fmaf(cm, cm, part);
      }

      ap += vstep * 8;                       // 16*gridDim rows of 8 float2
      ep += vstep * DIM;
    }
    atomicAdd(&lds_sumsq[d0 + l16], part);   // ds_add_f32, once per dtile
  }

  __syncthreads();
  for (int d = tid; d < DIM; d += 256)
    atomicAdd(&sumsq[d], lds_sumsq[d]);      // one global atomic per column/block
}

// ---------------------------------------------------------------------------
// Phase 2: add tail row(s) [NVF*16, VOCAB), then
//          norm_mag[d] = mag[d] / max(sqrt(sumsq[d]), 1e-8)
// ---------------------------------------------------------------------------
__global__ void dora_norm_finalize(const float* __restrict__ sumsq,
                                   const float* __restrict__ emb,
                                   const float* __restrict__ la,
                                   const float* __restrict__ lb,
                                   const float* __restrict__ mag,
                                   float* __restrict__ norm_mag)
{
  const int d = blockIdx.x * blockDim.x + threadIdx.x;
  if (d < DIM) {
    float s = sumsq[d];
    for (int v = NVF * 16; v < VOCAB; ++v) {   // exactly 1 row (50256)
      float acc = emb[(long)v * DIM + d];
#pragma unroll
      for (int k = 0; k < RANK; ++k)
        acc = fmaf(SCALING * la[(long)v * RANK + k], lb[k * DIM + d], acc);
      s = fmaf(acc, acc, s);
    }
    const float n = fmaxf(sqrtf(s), 1e-8f);
    norm_mag[d] = mag[d] / n;
  }
}

// ---------------------------------------------------------------------------
// Phase 3: out[t, d] = (emb[idx[t], d] + SCALING*(A@B)[idx[t], d]) * norm_mag[d]
// One wave per 16-token x 16-col tile. M = tokens, K = rank(16), N = cols.
// ---------------------------------------------------------------------------
__global__ void __launch_bounds__(256) dora_gather_kernel(
    const int*   __restrict__ idx,       // [16384]
    const float* __restrict__ emb,
    const float* __restrict__ la,
    const float* __restrict__ lb,
    const float* __restrict__ norm_mag,
    float* __restrict__ out)             // [16384, DIM]
{
  const int tid  = threadIdx.x;
  const int lane = tid & 31;
  const int wave = tid >> 5;
  const int half = lane >> 4;
  const int l16  = lane & 15;

  const int tileId = blockIdx.x * 8 + wave;      // 1024*48 tiles, 8 waves/block
  const int ttile  = tileId / 48;
  const int dtile  = tileId % 48;
  const int t0 = ttile * 16;
  const int d0 = dtile * 16;

  // Token row index for A-matrix row M = l16 (same in both lane halves).
  const int my_idx = idx[t0 + l16];

  const float2* la2 = reinterpret_cast<const float2*>(la);
  v2f a[4], b[4];
#pragma unroll
  for (int j = 0; j < 4; ++j) {
    const float2 t = la2[(long)my_idx * 8 + 2 * j + half];
    a[j][0] = t.x; a[j][1] = t.y;
    const int k = 4 * j + 2 * half;
    b[j][0] = lb[(k    ) * DIM + d0 + l16];
    b[j][1] = lb[(k + 1) * DIM + d0 + l16];
  }

  v8f c = {};
#pragma unroll
  for (int j = 0; j < 4; ++j)
    c = __builtin_amdgcn_wmma_f32_16x16x4_f32(
            false, a[j], false, b[j], (short)0, c, false, false);

  const float nm = norm_mag[d0 + l16];

  // Batch: 8 bpermutes, then 8 independent gather loads, then 8 NT stores.
  int im[8];
#pragma unroll
  for (int r = 0; r < 8; ++r)
    im[r] = __shfl(my_idx, r + 8 * half, 32);

  float e[8];
#pragma unroll
  for (int r = 0; r < 8; ++r)
    e[r] = emb[(long)im[r] * DIM + d0 + l16];     // coalesced 64B segments, L2-hot

  float* op = out + (long)(t0 + 8 * half) * DIM + d0 + l16;
#pragma unroll
  for (int r = 0; r < 8; ++r) {
    const float val = fmaf(SCALING, c[r], e[r]) * nm;
    __builtin_nontemporal_store(val, op + r * DIM);  // don't evict emb from L2
  }
}

// ---------------------------------------------------------------------------
extern "C" void kernel_launch(void* const* d_in, const int* in_sizes, int n_in,
                              void* d_out, int out_size, void* d_ws, size_t ws_size,
                              hipStream_t stream)
{
  const int*   inputs = (const int*)  d_in[0];   // [8, 2048]
  const float* emb    = (const float*)d_in[1];   // [VOCAB, DIM]
  const float* la     = (const float*)d_in[2];   // [VOCAB, RANK]
  const float* lb     = (const float*)d_in[3];   // [RANK, DIM]
  const float* mag    = (const float*)d_in[4];   // [DIM]
  float* out = (float*)d_out;

  float* sumsq    = (float*)d_ws;        // [DIM]
  float* norm_mag = sumsq + DIM;         // [DIM]

  // Workspace is poisoned once and never re-poisoned: zero the accumulator
  // every call (deterministic, graph-capture safe).
  hipMemsetAsync(sumsq, 0, DIM * sizeof(float), stream);

  dora_colnorm_kernel<<<256, 256, 0, stream>>>(emb, la, lb, sumsq);
  dora_norm_finalize<<<(DIM + 255) / 256, 256, 0, stream>>>(
      sumsq, emb, la, lb, mag, norm_mag);

  const int ntok   = in_sizes[0];          // 16384, multiple of 16
  const int ttiles = ntok / 16;            // 1024
  const int tiles  = ttiles * (DIM / 16);  // 49152, 8 waves per block
  dora_gather_kernel<<<tiles / 8, 256, 0, stream>>>(
      inputs, emb, la, lb, norm_mag, out);
}